// MHCLayer_44023414784003
// MI455X (gfx1250) — compile-verified
//
#include <hip/hip_runtime.h>
#include <hip/hip_bf16.h>

typedef __attribute__((ext_vector_type(2))) float v2f;
typedef __attribute__((ext_vector_type(4))) float v4f;
typedef __attribute__((ext_vector_type(8))) float v8f;

#define BSZ   4096
#define NS    4
#define CDIM  4096
#define NC    16384      // NS * CDIM
#define KPAD  32         // padded projection count (24 used)
#define ROWS  16         // batch rows per block (WMMA M)
#define THREADS 256
#define NWAVES 8
#define CPW   (CDIM / NWAVES)   // 512 columns per wave

// ---------------------------------------------------------------------------
// WMMA: D(16x16 f32) = A(16x4 f32) * B(4x16 f32) + C
// A: 2 VGPR (v2f): VGPR0 = K0 (lanes 0-15) / K2 (lanes 16-31); VGPR1 = K1/K3
// B: 2 VGPR (v2f): same K striping, N = lane&15
// C/D: 8 VGPR (v8f): M = slot + 8*(lane>=16), N = lane&15
// ---------------------------------------------------------------------------
static __device__ inline v8f wmma16x16x4(v2f a, v2f b, v8f c) {
#if defined(__gfx1250__) && __has_builtin(__builtin_amdgcn_wmma_f32_16x16x4_f32)
  return __builtin_amdgcn_wmma_f32_16x16x4_f32(false, a, false, b, (short)0, c,
                                               false, false);
#else
  c[0] += a.x * b.x + a.y * b.y;   // compile-safety fallback (host pass)
  return c;
#endif
}

// ---------------------------------------------------------------------------
// Prep: W3[ ((c*2 + h)*32 + q)*2 + j ] = phi_all[q][ (2h+j)*CDIM + c ]
// (q = phi output 0..31, 24 used: rows 0-3 phi_pre, 4-7 phi_post, 8-23 phi_res;
//  h = half-wave, j selects K / K+1). This makes each lane's B-fragment one
//  contiguous b64 load.  Size: 4096*2*32*2 floats = 2 MB (L2-resident).
// ---------------------------------------------------------------------------
__global__ __launch_bounds__(256) void prep_w3(const float* __restrict__ phi_pre,
                                               const float* __restrict__ phi_post,
                                               const float* __restrict__ phi_res,
                                               float* __restrict__ W3) {
  int idx = blockIdx.x * 256 + threadIdx.x;
  if (idx >= CDIM * 2 * KPAD * 2) return;
  int j = idx & 1;
  int q = (idx >> 1) & 31;
  int h = (idx >> 6) & 1;
  int c = idx >> 7;
  int n = 2 * h + j;              // K index within the 4-stream group
  int t = n * CDIM + c;           // flat column in phi
  float v = 0.0f;
  if (q < 4)       v = phi_pre[q * NC + t];
  else if (q < 8)  v = phi_post[(q - 4) * NC + t];
  else if (q < 24) v = phi_res[(q - 8) * NC + t];
  W3[idx] = v;
}

// ---------------------------------------------------------------------------
// Fused main kernel: one block = 16 batch rows.
// Phase 1: WMMA projections d[16,24] + per-row 4x4 Gram G (same x loads).
// Phase 2: per-row scalar math (softmax / sigmoid / exp / 20x Sinkhorn / RMS).
// Phase 3: stream x again (L2-hot) -> out = Hpost*ynorm + M@x  (b128 ld/st).
// ---------------------------------------------------------------------------
__global__ __launch_bounds__(THREADS) void mhc_main(
    const float* __restrict__ x, const float* __restrict__ wgt,
    const float* __restrict__ W3,
    const float* __restrict__ b_pre, const float* __restrict__ b_post,
    const float* __restrict__ b_res,
    const float* __restrict__ a_pre, const float* __restrict__ a_post,
    const float* __restrict__ a_res,
    float* __restrict__ out) {
  __shared__ float redD[NWAVES][2][32][8];   // 16 KB partial WMMA tiles
  __shared__ float dmat[ROWS][KPAD];
  __shared__ float sG[ROWS][16];
  __shared__ float sHpre[ROWS][4];
  __shared__ float sHpost[ROWS][4];
  __shared__ float sM[ROWS][16];
  __shared__ float sRs[ROWS];

  const int tid  = threadIdx.x;
  const int wave = tid >> 5;
  const int lane = tid & 31;
  const int row  = lane & 15;
  const int half = lane >> 4;
  const int b0   = blockIdx.x * ROWS;

  ((float*)sG)[tid] = 0.0f;   // 16*16 == THREADS
  __syncthreads();

  // ---------------- Phase 1: projections (WMMA) + Gram ----------------
  v8f acc0 = {};   // phi outputs 0..15
  v8f acc1 = {};   // phi outputs 16..23 (24..31 zero-padded)
  v2f gv[16];      // packed Gram partials (reduce .x+.y at end)
#pragma unroll
  for (int i = 0; i < 16; ++i) gv[i] = (v2f){0.0f, 0.0f};

  const float* xrow = x + (size_t)(b0 + row) * NC;
  const int nb = lane & 15;
  const int c_begin = wave * CPW;
  const int c_end   = c_begin + CPW;

  for (int c = c_begin; c < c_end; c += 2) {
    // 4 streams x 2 columns: b64 loads
    v2f xp[4];
#pragma unroll
    for (int n = 0; n < 4; ++n) xp[n] = *(const v2f*)(xrow + n * CDIM + c);

    // A fragments for columns c and c+1 (branch-free half select)
    v2f afA, afB;
    afA.x = half ? xp[2].x : xp[0].x;
    afA.y = half ? xp[3].x : xp[1].x;
    afB.x = half ? xp[2].y : xp[0].y;
    afB.y = half ? xp[3].y : xp[1].y;

    // B fragments: one b64 per (tile, column)
    const float* w3A = W3 + (((size_t)c * 2 + half) * KPAD) * 2;
    const float* w3B = w3A + 2 * KPAD * 2;   // column c+1
    v2f bf0A = *(const v2f*)(w3A + nb * 2);
    v2f bf1A = *(const v2f*)(w3A + (nb + 16) * 2);
    v2f bf0B = *(const v2f*)(w3B + nb * 2);
    v2f bf1B = *(const v2f*)(w3B + (nb + 16) * 2);

    acc0 = wmma16x16x4(afA, bf0A, acc0);
    acc1 = wmma16x16x4(afA, bf1A, acc1);
    acc0 = wmma16x16x4(afB, bf0B, acc0);
    acc1 = wmma16x16x4(afB, bf1B, acc1);

    // Gram (all lanes, packed over the column pair -> v_pk_fma_f32)
#pragma unroll
    for (int n = 0; n < 4; ++n)
#pragma unroll
      for (int m = 0; m < 4; ++m) gv[n * 4 + m] += xp[n] * xp[m];
  }

  // spill partial WMMA accumulators; commit Gram from lower half-wave only
#pragma unroll
  for (int i = 0; i < 8; ++i) {
    redD[wave][0][lane][i] = acc0[i];
    redD[wave][1][lane][i] = acc1[i];
  }
  if (!half) {
#pragma unroll
    for (int i = 0; i < 16; ++i) atomicAdd(&sG[row][i], gv[i].x + gv[i].y);
  }
  __syncthreads();

  // tree-reduce 8 waves' tiles -> dmat[m][N]
  for (int p = tid; p < 512; p += THREADS) {
    int tile = p >> 8;
    int rem  = p & 255;
    int ln   = rem >> 3;
    int slot = rem & 7;
    float s = 0.0f;
#pragma unroll
    for (int wv = 0; wv < NWAVES; ++wv) s += redD[wv][tile][ln][slot];
    int m = slot + ((ln >> 4) << 3);
    int N = (ln & 15) + tile * 16;
    dmat[m][N] = s;
  }
  __syncthreads();

  // ---------------- Phase 2: per-row scalar math ----------------
  if (tid < ROWS) {
    const int r = tid;
    float ss = sG[r][0] + sG[r][5] + sG[r][10] + sG[r][15];  // trace(G)
    float s  = rsqrtf(ss * (1.0f / (float)NC) + 1e-8f);
    float ap = a_pre[0], aq = a_post[0], ar = a_res[0];

    float hpre[4], hpost[4], Mm[4][4];
#pragma unroll
    for (int k = 0; k < 4; ++k) {
      hpre[k]  = ap * s * dmat[r][k] + b_pre[k];
      hpost[k] = aq * s * dmat[r][4 + k] + b_post[k];
    }
#pragma unroll
    for (int i = 0; i < 4; ++i)
#pragma unroll
      for (int j = 0; j < 4; ++j)
        Mm[i][j] = ar * s * dmat[r][8 + i * 4 + j] + b_res[i * 4 + j];

    float mx = fmaxf(fmaxf(hpre[0], hpre[1]), fmaxf(hpre[2], hpre[3]));
    float e[4], se = 0.0f;
#pragma unroll
    for (int k = 0; k < 4; ++k) { e[k] = __expf(hpre[k] - mx); se += e[k]; }
    float Hp[4];
#pragma unroll
    for (int k = 0; k < 4; ++k) Hp[k] = e[k] / se;

    float Hq[4];
#pragma unroll
    for (int k = 0; k < 4; ++k) Hq[k] = 2.0f / (1.0f + __expf(-hpost[k]));

#pragma unroll
    for (int i = 0; i < 4; ++i)
#pragma unroll
      for (int j = 0; j < 4; ++j) Mm[i][j] = __expf(Mm[i][j]);
    for (int it = 0; it < 20; ++it) {
#pragma unroll
      for (int i = 0; i < 4; ++i) {
        float inv = 1.0f / (Mm[i][0] + Mm[i][1] + Mm[i][2] + Mm[i][3] + 1e-8f);
#pragma unroll
        for (int j = 0; j < 4; ++j) Mm[i][j] *= inv;
      }
#pragma unroll
      for (int j = 0; j < 4; ++j) {
        float inv = 1.0f / (Mm[0][j] + Mm[1][j] + Mm[2][j] + Mm[3][j] + 1e-8f);
#pragma unroll
        for (int i = 0; i < 4; ++i) Mm[i][j] *= inv;
      }
    }

    float sa = 0.0f;   // sum_c x_agg^2 = Hp^T G Hp
#pragma unroll
    for (int n = 0; n < 4; ++n)
#pragma unroll
      for (int m = 0; m < 4; ++m) sa += Hp[n] * Hp[m] * sG[r][n * 4 + m];
    sRs[r] = rsqrtf(sa * (1.0f / (float)CDIM) + 1e-5f);

#pragma unroll
    for (int k = 0; k < 4; ++k) { sHpre[r][k] = Hp[k]; sHpost[r][k] = Hq[k]; }
#pragma unroll
    for (int i = 0; i < 16; ++i) sM[r][i] = Mm[i >> 2][i & 3];
  }
  __syncthreads();

  // ---------------- Phase 3: out = Hpost*ynorm + M @ x (b128 traffic) -------
  for (int r = 0; r < ROWS; ++r) {
    const float* xb = x + (size_t)(b0 + r) * NC;
    float* ob = out + (size_t)(b0 + r) * NC;
    const float hp0 = sHpre[r][0], hp1 = sHpre[r][1];
    const float hp2 = sHpre[r][2], hp3 = sHpre[r][3];
    const float hq0 = sHpost[r][0], hq1 = sHpost[r][1];
    const float hq2 = sHpost[r][2], hq3 = sHpost[r][3];
    const float rsO = sRs[r];
    float Mr[16];
#pragma unroll
    for (int i = 0; i < 16; ++i) Mr[i] = sM[r][i];

    for (int c0 = tid * 4; c0 < CDIM; c0 += THREADS * 4) {
      v4f x0 = *(const v4f*)(xb + c0);
      v4f x1 = *(const v4f*)(xb + CDIM + c0);
      v4f x2 = *(const v4f*)(xb + 2 * CDIM + c0);
      v4f x3 = *(const v4f*)(xb + 3 * CDIM + c0);
      v4f wv = *(const v4f*)(wgt + c0);
      v4f xa = hp0 * x0 + hp1 * x1 + hp2 * x2 + hp3 * x3;
      v4f yn = xa * rsO * wv;
      *(v4f*)(ob + c0)            = hq0 * yn + Mr[0]  * x0 + Mr[1]  * x1 + Mr[2]  * x2 + Mr[3]  * x3;
      *(v4f*)(ob + CDIM + c0)     = hq1 * yn + Mr[4]  * x0 + Mr[5]  * x1 + Mr[6]  * x2 + Mr[7]  * x3;
      *(v4f*)(ob + 2 * CDIM + c0) = hq2 * yn + Mr[8]  * x0 + Mr[9]  * x1 + Mr[10] * x2 + Mr[11] * x3;
      *(v4f*)(ob + 3 * CDIM + c0) = hq3 * yn + Mr[12] * x0 + Mr[13] * x1 + Mr[14] * x2 + Mr[15] * x3;
    }
  }
}

extern "C" void kernel_launch(void* const* d_in, const int* in_sizes, int n_in,
                              void* d_out, int out_size, void* d_ws, size_t ws_size,
                              hipStream_t stream) {
  const float* x        = (const float*)d_in[0];
  const float* wgt      = (const float*)d_in[1];
  const float* phi_pre  = (const float*)d_in[2];
  const float* phi_post = (const float*)d_in[3];
  const float* phi_res  = (const float*)d_in[4];
  const float* b_pre    = (const float*)d_in[5];
  const float* b_post   = (const float*)d_in[6];
  const float* b_res    = (const float*)d_in[7];
  const float* a_pre    = (const float*)d_in[8];
  const float* a_post   = (const float*)d_in[9];
  const float* a_res    = (const float*)d_in[10];
  float* out = (float*)d_out;
  float* W3  = (float*)d_ws;   // 2 MB

  prep_w3<<<(CDIM * 2 * KPAD * 2 + 255) / 256, 256, 0, stream>>>(
      phi_pre, phi_post, phi_res, W3);
  mhc_main<<<BSZ / ROWS, THREADS, 0, stream>>>(x, wgt, W3, b_pre, b_post, b_res,
                                               a_pre, a_post, a_res, out);
}